// VectorQuantizerEMA_65532611002587
// MI455X (gfx1250) — compile-verified
//
#include <hip/hip_runtime.h>
#include <math.h>

// ---------------------------------------------------------------------------
// VQ-VAE eval forward for MI455X (gfx1250, wave32).
//   inputs:    (B*T, 64) f32, N = 65536 rows
//   embedding: (K, 64)   f32, K = 1024 codes
// d_out (f32): [0..N*64) quantized_st | [N*64] loss | [N*64+1] perplexity
//
// Distance GEMM: V_WMMA_F32_16X16X4_F32 (exact fp32 — argmin is
// discontinuous). Codebook is double-buffer staged into LDS with
// GLOBAL_LOAD_ASYNC_TO_LDS_B128 (ASYNCcnt path), shared by 8 waves,
// and each wave computes 2 row-tiles so every LDS B-fragment feeds two
// independent WMMA accumulation chains.
// ---------------------------------------------------------------------------

typedef __attribute__((ext_vector_type(2))) float v2f;
typedef __attribute__((ext_vector_type(8))) float v8f;
typedef int v4i __attribute__((ext_vector_type(4)));
typedef __attribute__((address_space(1))) v4i gv4i;   // global v4i
typedef __attribute__((address_space(3))) v4i lv4i;   // LDS v4i

#define VQ_D 64
#define VQ_KSTEPS 16            // D/4 k-steps of the 16x16x4 f32 WMMA
#define WAVES 8                 // 256 threads
#define ROWS_PER_WAVE 32        // two 16-row tiles per wave
#define ROWS_PER_BLOCK 256      // WAVES * ROWS_PER_WAVE
#define CHUNK_CODES 64          // codes staged to LDS per round (16 KB)
#define LDS_STRIDE 68           // 64 + 4 dword pad: 16B-aligned rows, 0-conflict

// --- async global->LDS 16B copy (CDNA5 ASYNCcnt path) -----------------------
__device__ __forceinline__ void async_ld16(const float* g, float* l) {
#if __has_builtin(__builtin_amdgcn_global_load_async_to_lds_b128)
    // signature (from clang diagnostic): (v4i AS1*, v4i AS3*, imm offset, imm cpol)
    __builtin_amdgcn_global_load_async_to_lds_b128(
        (gv4i*)g, (lv4i*)l, 0, 0);
#else
    unsigned lofs = (unsigned)(size_t)(lv4i*)l;
    asm volatile("global_load_async_to_lds_b128 %0, %1, off"
                 :: "v"(lofs), "v"(g) : "memory");
#endif
}

__device__ __forceinline__ void wait_async0() {
#if __has_builtin(__builtin_amdgcn_s_wait_asynccnt)
    __builtin_amdgcn_s_wait_asynccnt(0);
#else
    asm volatile("s_wait_asynccnt 0x0" ::: "memory");
#endif
}

// Wave `wave` stages its 8 code rows of chunk `chunk` into `buf`.
// 8 rows x 64 floats = 128 x 16B units = 32 lanes x 4 async B128 ops.
__device__ __forceinline__ void stage_chunk(const float* __restrict__ emb,
                                            float* buf, int chunk,
                                            int wave, int lane) {
    const float* g0 = emb + ((size_t)chunk * CHUNK_CODES + wave * 8) * VQ_D;
    float* l0 = buf + (wave * 8) * LDS_STRIDE;
#pragma unroll
    for (int j = 0; j < 4; ++j) {
        int u   = j * 32 + lane;   // 0..127
        int r   = u >> 4;          // code row 0..7
        int seg = u & 15;          // 16B segment within row
        async_ld16(g0 + r * VQ_D + seg * 4, l0 + r * LDS_STRIDE + seg * 4);
    }
}

// --- Kernel A: codebook row norms + zero per-launch accumulators ------------
__global__ __launch_bounds__(256) void vq_prep(
    const float* __restrict__ emb, float* __restrict__ enorm,
    float* __restrict__ counts, float* __restrict__ lossAcc, int K)
{
    int k = blockIdx.x * blockDim.x + threadIdx.x;
    if (k < K) {
        const float4* e4 = (const float4*)(emb + (size_t)k * VQ_D);
        float s = 0.0f;
#pragma unroll
        for (int i = 0; i < VQ_D / 4; ++i) {
            float4 v = e4[i];
            s += v.x * v.x + v.y * v.y + v.z * v.z + v.w * v.w;
        }
        enorm[k]  = s;
        counts[k] = 0.0f;
    }
    if (k == 0) lossAcc[0] = 0.0f;
}

// --- Kernel B: argmin over K codes, WMMA from LDS-staged codebook -----------
// Requires N % ROWS_PER_BLOCK == 0 (true for B*T = 65536).
__global__ __launch_bounds__(256) void vq_argmin_wmma(
    const float* __restrict__ inp, const float* __restrict__ emb,
    const float* __restrict__ enorm, int* __restrict__ idxbuf, int K)
{
    __shared__ float ldsb[2][CHUNK_CODES * LDS_STRIDE];   // 2 x 17 KB

    const int lane = threadIdx.x & 31;
    const int wave = threadIdx.x >> 5;
    const int half = lane >> 4;          // 0: k-pair {4s,4s+1}, 1: {4s+2,4s+3}
    const int lr   = lane & 15;          // row (A) / column (B,C) in tile
    const int rowBase = blockIdx.x * ROWS_PER_BLOCK + wave * ROWS_PER_WAVE;

    // Preload this wave's two A tiles (rows rowBase.. +15 and +16.. +31).
    v2f a0[VQ_KSTEPS], a1[VQ_KSTEPS];
    {
        const float* ar0 = inp + (size_t)(rowBase + lr) * VQ_D + 2 * half;
        const float* ar1 = ar0 + 16 * VQ_D;
#pragma unroll
        for (int s = 0; s < VQ_KSTEPS; ++s) {
            a0[s] = *(const v2f*)(ar0 + 4 * s);
            a1[s] = *(const v2f*)(ar1 + 4 * s);
        }
    }

    float minv0[8], minv1[8];
    int   mini0[8], mini1[8];
#pragma unroll
    for (int v = 0; v < 8; ++v) {
        minv0[v] = 3.402823466e38f; mini0[v] = 0;
        minv1[v] = 3.402823466e38f; mini1[v] = 0;
    }

    stage_chunk(emb, ldsb[0], 0, wave, lane);   // prologue: chunk 0 in flight

    const int nchunks = K / CHUNK_CODES;
    for (int ch = 0; ch < nchunks; ++ch) {
        wait_async0();          // my chunk-ch async loads have landed
        __syncthreads();        // everyone's landed; prev compute done (WAR ok)
        if (ch + 1 < nchunks)   // overlap next chunk with this compute
            stage_chunk(emb, ldsb[(ch + 1) & 1], ch + 1, wave, lane);

        const float* bbuf = ldsb[ch & 1];
#pragma unroll
        for (int ct = 0; ct < CHUNK_CODES / 16; ++ct) {
            const int codeLocal = ct * 16 + lr;
            const float* brow = bbuf + codeLocal * LDS_STRIDE + 2 * half;

            v8f acc0 = {}, acc1 = {};
#pragma unroll
            for (int s = 0; s < VQ_KSTEPS; ++s) {
                v2f b = *(const v2f*)(brow + 4 * s);   // ds_load_b64, shared
                acc0 = __builtin_amdgcn_wmma_f32_16x16x4_f32(
                    false, a0[s], false, b, (short)0, acc0, false, false);
                acc1 = __builtin_amdgcn_wmma_f32_16x16x4_f32(
                    false, a1[s], false, b, (short)0, acc1, false, false);
            }

            const int   code = ch * CHUNK_CODES + codeLocal;
            const float en   = enorm[code];            // ||e_code||^2
#pragma unroll
            for (int v = 0; v < 8; ++v) {
                float d0 = fmaf(-2.0f, acc0[v], en);   // row-norm dropped
                float d1 = fmaf(-2.0f, acc1[v], en);
                if (d0 < minv0[v]) { minv0[v] = d0; mini0[v] = code; }
                if (d1 < minv1[v]) { minv1[v] = d1; mini1[v] = code; }
            }
        }
    }

    // Reduce across the 16 lanes sharing each row; tie-break on smaller index
    // to match jnp.argmin.
#pragma unroll
    for (int v = 0; v < 8; ++v) {
#pragma unroll
        for (int off = 8; off >= 1; off >>= 1) {
            float ov0 = __shfl_xor(minv0[v], off, 16);
            int   oi0 = __shfl_xor(mini0[v], off, 16);
            if (ov0 < minv0[v] || (ov0 == minv0[v] && oi0 < mini0[v])) {
                minv0[v] = ov0; mini0[v] = oi0;
            }
            float ov1 = __shfl_xor(minv1[v], off, 16);
            int   oi1 = __shfl_xor(mini1[v], off, 16);
            if (ov1 < minv1[v] || (ov1 == minv1[v] && oi1 < mini1[v])) {
                minv1[v] = ov1; mini1[v] = oi1;
            }
        }
    }

    if (lr == 0) {
#pragma unroll
        for (int v = 0; v < 8; ++v) {                 // row M = v + 8*half
            idxbuf[rowBase +      half * 8 + v] = mini0[v];
            idxbuf[rowBase + 16 + half * 8 + v] = mini1[v];
        }
    }
}

// --- Kernel C: gather quantized, straight-through output, loss + histogram --
__global__ __launch_bounds__(256) void vq_gather(
    const float* __restrict__ inp, const float* __restrict__ emb,
    const int* __restrict__ idxbuf, float* __restrict__ out,
    float* __restrict__ counts, float* __restrict__ lossAcc, int total)
{
    const int e   = blockIdx.x * 256 + threadIdx.x;
    const int tid = threadIdx.x;
    float diff2 = 0.0f;
    if (e < total) {
        const int row = e >> 6;
        const int d   = e & 63;
        const int idx = idxbuf[row];
        const float q = emb[(size_t)idx * VQ_D + d];
        const float z = inp[e];
        out[e] = z + (q - z);                // quantized_st, bit-exact STE form
        const float diff = z - q;
        diff2 = diff * diff;
        if (d == 0) atomicAdd(&counts[idx], 1.0f);
    }
    __shared__ float sdata[256];
    sdata[tid] = diff2;
    __syncthreads();
#pragma unroll
    for (int s = 128; s > 0; s >>= 1) {
        if (tid < s) sdata[tid] += sdata[tid + s];
        __syncthreads();
    }
    if (tid == 0) atomicAdd(lossAcc, sdata[0]);
}

// --- Kernel D: finalize loss + perplexity -----------------------------------
__global__ __launch_bounds__(256) void vq_finalize(
    const float* __restrict__ counts, const float* __restrict__ lossAcc,
    float* __restrict__ out, int K, int N, int total)
{
    __shared__ float sdata[256];
    const int tid = threadIdx.x;
    const float invN = 1.0f / (float)N;
    float local = 0.0f;
    for (int k = tid; k < K; k += 256) {
        float p = counts[k] * invN;
        local += p * logf(p + 1e-10f);
    }
    sdata[tid] = local;
    __syncthreads();
#pragma unroll
    for (int s = 128; s > 0; s >>= 1) {
        if (tid < s) sdata[tid] += sdata[tid + s];
        __syncthreads();
    }
    if (tid == 0) {
        float entropy = sdata[0];
        float perp = expf(-entropy);
        perp = fminf(perp, (float)K);
        if (!isfinite(perp)) perp = 0.0f;
        out[total]     = 0.25f * lossAcc[0] / (float)total;   // commitment loss
        out[total + 1] = perp;
    }
}

// ---------------------------------------------------------------------------
extern "C" void kernel_launch(void* const* d_in, const int* in_sizes, int n_in,
                              void* d_out, int out_size, void* d_ws, size_t ws_size,
                              hipStream_t stream) {
    const float* inp = (const float*)d_in[0];   // (B*T, 64) f32
    const float* emb = (const float*)d_in[1];   // (K, 64)   f32
    float* out = (float*)d_out;

    const int total = in_sizes[0];              // N * 64
    const int N     = total / VQ_D;
    const int K     = in_sizes[1] / VQ_D;

    // Workspace layout (fully rewritten each launch — graph-replay safe):
    char*  ws      = (char*)d_ws;
    float* enorm   = (float*)ws;                                        // K
    float* counts  = enorm + K;                                         // K
    float* lossAcc = counts + K;                                        // 1
    int*   idxbuf  = (int*)(ws + (size_t)(2 * K + 64) * sizeof(float)); // N

    vq_prep<<<(K + 255) / 256, 256, 0, stream>>>(emb, enorm, counts, lossAcc, K);

    vq_argmin_wmma<<<N / ROWS_PER_BLOCK, 256, 0, stream>>>(inp, emb, enorm,
                                                           idxbuf, K);

    vq_gather<<<(total + 255) / 256, 256, 0, stream>>>(inp, emb, idxbuf, out,
                                                       counts, lossAcc, total);

    vq_finalize<<<1, 256, 0, stream>>>(counts, lossAcc, out, K, N, total);
}